// QueryAndGroup_14688788152829
// MI455X (gfx1250) — compile-verified
//
#include <hip/hip_runtime.h>

typedef __attribute__((ext_vector_type(2))) float v2f;
typedef __attribute__((ext_vector_type(8))) float v8f;

#define BQ_B 4
#define BQ_N 8192
#define BQ_M 4096
#define BQ_C 64
#define BQ_S 32
#define BQ_R2 (0.12f * 0.12f)
#define BQ_R2_PRE (BQ_R2 + 1.0e-4f)   // WMMA prefilter margin (exact check follows)

// One wave handles 16 query points. Distance tile (16 queries x 16 points)
// computed with V_WMMA_F32_16X16X4_F32:
//   A[m,:] = (-2qx, -2qy, -2qz, |q|^2)   (16x4, lanes 0-15: K0,K1; 16-31: K2,K3)
//   B[:,n] = ( px,  py,  pz,  1 )        (4x16, same K split)
//   C[m,n] = |p_n|^2                     (lane holds column n = lane&15)
// => D[m,n] = |q_m - p_n|^2 (expanded form). Used as prefilter; candidates are
// re-verified with exact (q-p)^2 to match reference rounding.
__global__ __launch_bounds__(256) void ball_query_wmma_kernel(
    const float* __restrict__ xyz, const float* __restrict__ new_xyz,
    int* __restrict__ idx_out)
{
    const int lane  = threadIdx.x & 31;
    const int wave  = threadIdx.x >> 5;
    const int g     = blockIdx.x * 8 + wave;      // 0 .. B*M/16 - 1 (exact)
    const int b     = g >> 8;                     // M/16 = 256 groups per batch
    const int q0    = (g & 255) << 4;
    const int col   = lane & 15;
    const int khalf = lane >> 4;

    // ---- Load my query (row col of the tile) and build A ----
    const float* qp = new_xyz + (size_t)(b * BQ_M + q0 + col) * 3;
    const float qx = qp[0], qy = qp[1], qz = qp[2];
    const float qn2 = qx * qx + qy * qy + qz * qz;

    v2f amat;
    amat.x = khalf ? (-2.0f * qz) : (-2.0f * qx);   // K2 : K0
    amat.y = khalf ? qn2          : (-2.0f * qy);   // K3 : K1

    const float* xb = xyz + (size_t)b * BQ_N * 3;

    int  cnt = 0;
    int  firstIdx = 0;
    bool haveFirst = false;
    const int row = q0 + col;                       // valid query row for lanes 0-15
    int* my_out = idx_out + (size_t)(b * BQ_M + row) * BQ_S;

    for (int tile = 0; tile < BQ_N / 16; ++tile) {
        // ---- Load point column (n = tile*16 + col), build B and C ----
        const float* pp = xb + (size_t)(tile * 16 + col) * 3;
        const float px = pp[0], py = pp[1], pz = pp[2];
        const float pn2 = px * px + py * py + pz * pz;

        v2f bmat;
        bmat.x = khalf ? pz   : px;                 // K2 : K0
        bmat.y = khalf ? 1.0f : py;                 // K3 : K1

        v8f cacc;
#pragma unroll
        for (int i = 0; i < 8; ++i) cacc[i] = pn2;

        // D = A*B + C  (full-EXEC here: every lane in every wave is live)
        v8f d = __builtin_amdgcn_wmma_f32_16x16x4_f32(
            false, amat, false, bmat, (short)0, cacc, false, false);

        // ---- Per-row candidate masks: D vgpr i holds rows i (lanes 0-15,
        // bits 0-15 = cols) and i+8 (lanes 16-31, bits 16-31 = cols) ----
        unsigned bits = 0;
#pragma unroll
        for (int i = 0; i < 8; ++i) {
            unsigned mi = __builtin_amdgcn_ballot_w32(d[i] < BQ_R2_PRE);
            if (lane < 16 && (lane & 7) == i)
                bits = (lane < 8) ? (mi & 0xFFFFu) : (mi >> 16);
        }

        // ---- Lanes 0-15: append candidates in ascending index order, with
        // exact distance re-check (matches reference (q-p)^2 rounding) ----
        if (lane < 16) {
            const int base = tile * 16;
            while (bits && cnt < BQ_S) {
                const int n = base + __builtin_ctz(bits);
                bits &= bits - 1;
                const float* cp = xb + (size_t)n * 3;
                const float dx = qx - cp[0];
                const float dy = qy - cp[1];
                const float dz = qz - cp[2];
                const float d2 = dx * dx + dy * dy + dz * dz;
                if (d2 < BQ_R2) {
                    if (!haveFirst) { firstIdx = n; haveFirst = true; }
                    my_out[cnt++] = n;
                }
            }
        }

        // ---- Early exit once all 16 rows are full (wave-uniform) ----
        const unsigned alive =
            __builtin_amdgcn_ballot_w32(lane < 16 && cnt < BQ_S);
        if (alive == 0) break;
    }

    // ---- Pad remaining slots with first hit (0 if no hit, matching top_k tie) ----
    if (lane < 16) {
        const int pad = haveFirst ? firstIdx : 0;
        for (int s = cnt; s < BQ_S; ++s) my_out[s] = pad;
    }
}

// Gather + center + concat. grid = (M*S/256, 3+C, B); a wave covers one
// (b,c,m) row of 32 samples -> coalesced idx loads and output stores.
__global__ __launch_bounds__(256) void group_concat_kernel(
    const float* __restrict__ xyz, const float* __restrict__ new_xyz,
    const float* __restrict__ feat, const int* __restrict__ idx,
    float* __restrict__ out)
{
    const int c = blockIdx.y;
    const int b = blockIdx.z;
    const int t = blockIdx.x * 256 + threadIdx.x;   // over M*S = 131072
    const int s = t & (BQ_S - 1);
    const int m = t >> 5;

    const int n = idx[(size_t)(b * BQ_M + m) * BQ_S + s];

    float val;
    if (c < 3) {
        val = xyz[(size_t)(b * BQ_N + n) * 3 + c]
            - new_xyz[(size_t)(b * BQ_M + m) * 3 + c];
    } else {
        val = feat[((size_t)(b * BQ_C + (c - 3))) * BQ_N + n];
    }
    out[(((size_t)(b * (3 + BQ_C) + c)) * BQ_M + m) * BQ_S + s] = val;
}

extern "C" void kernel_launch(void* const* d_in, const int* in_sizes, int n_in,
                              void* d_out, int out_size, void* d_ws, size_t ws_size,
                              hipStream_t stream) {
    const float* xyz     = (const float*)d_in[0];   // (B,N,3)
    const float* new_xyz = (const float*)d_in[1];   // (B,M,3)
    const float* feat    = (const float*)d_in[2];   // (B,C,N)
    float* out = (float*)d_out;                     // (B,3+C,M,S)
    int*   idx = (int*)d_ws;                        // (B,M,S) int32, 2 MB

    // Ball query: B*M/16 = 1024 waves -> 128 blocks of 8 waves.
    ball_query_wmma_kernel<<<(BQ_B * BQ_M / 16) / 8, 256, 0, stream>>>(
        xyz, new_xyz, idx);

    // Group + concat: (512, 67, 4) blocks of 256 threads.
    dim3 grid2(BQ_M * BQ_S / 256, 3 + BQ_C, BQ_B);
    group_concat_kernel<<<grid2, 256, 0, stream>>>(xyz, new_xyz, feat, idx, out);
}